// ContrastiveLoss_44530220925482
// MI455X (gfx1250) — compile-verified
//
#include <hip/hip_runtime.h>

// ---------------- problem constants ----------------
#define BATCH   50
#define NROWS   100          // 2*BATCH
#define DIM     524288
#define GROUP   5
#define TILE    16
#define NT      7            // ceil(100/16) -> padded 112
#define PADN    112
#define SPLITK  32           // K-splits (grid.y)
#define WPB     8            // waves per block
#define KBLOCK  (DIM / SPLITK)        // 16384 elements of K per block
#define KWAVE   (KBLOCK / WPB)        // 2048 elements of K per wave
#define KITER   (KWAVE / 16)          // 128 loop iterations, 4 WMMAs each

typedef __attribute__((ext_vector_type(2))) float v2f;
typedef __attribute__((ext_vector_type(8))) float v8f;

__device__ __forceinline__ const float* row_ptr(const float* ei, const float* ej, int r) {
    return (r < BATCH) ? (ei + (long)r * DIM) : (ej + (long)(r - BATCH) * DIM);
}

// -------- kernel 0: zero the padded gram accumulator --------
__global__ void zero_gram_kernel(float* gram) {
    int i = blockIdx.x * blockDim.x + threadIdx.x;
    if (i < PADN * PADN) gram[i] = 0.0f;
}

// -------- kernel 1: per-row 1/||row|| (on raw data) --------
__global__ __launch_bounds__(256) void norms_kernel(const float* __restrict__ ei,
                                                    const float* __restrict__ ej,
                                                    float* __restrict__ inv_norm) {
    __shared__ float red[256];
    const int row = blockIdx.x;
    const float* p = row_ptr(ei, ej, row);
    float ss = 0.0f;
    for (int i = threadIdx.x * 4; i < DIM; i += 256 * 4) {
        float4 v = *(const float4*)(p + i);
        ss += v.x * v.x + v.y * v.y + v.z * v.z + v.w * v.w;
    }
    red[threadIdx.x] = ss;
    __syncthreads();
    for (int s = 128; s > 0; s >>= 1) {
        if (threadIdx.x < s) red[threadIdx.x] += red[threadIdx.x + s];
        __syncthreads();
    }
    if (threadIdx.x == 0)
        inv_norm[row] = 1.0f / fmaxf(sqrtf(red[0]), 1e-12f);
}

// -------- kernel 2: split-K Gram via V_WMMA_F32_16X16X4_F32 --------
// grid = (49 tiles, SPLITK), block = 256 (8 waves). Each wave owns one
// 16x16 output tile over a 2048-long K chunk (512 WMMAs across 4
// independent accumulator chains), then atomically accumulates its 8
// C-values per lane into gram[112x112]. Padded rows (100..111) are
// clamped to row 0: their garbage lands only in the unread padded
// region of the workspace Gram, so no per-iteration masking is needed.
__global__ __launch_bounds__(256) void gram_kernel(const float* __restrict__ ei,
                                                   const float* __restrict__ ej,
                                                   float* __restrict__ gram) {
    const int lane = threadIdx.x & 31;
    const int wave = threadIdx.x >> 5;
    const int tile = blockIdx.x;                 // 0..48
    const int m0 = (tile / NT) * TILE;
    const int n0 = (tile % NT) * TILE;

    const int lrow = lane & 15;                  // row-in-tile for A/B fragments
    const int half = lane >> 4;                  // 0: K={0,1}  1: K={2,3}
    const int kOff = half * 2;

    const int rowA = m0 + lrow;
    const int rowB = n0 + lrow;
    const long kStart = (long)blockIdx.y * KBLOCK + (long)wave * KWAVE;
    const float* pA = row_ptr(ei, ej, (rowA < NROWS) ? rowA : 0) + kStart + kOff;
    const float* pB = row_ptr(ei, ej, (rowB < NROWS) ? rowB : 0) + kStart + kOff;

    v8f acc0 = {}, acc1 = {}, acc2 = {}, acc3 = {};
#pragma unroll 2
    for (int it = 0; it < KITER; ++it) {
        // A fragment (16x4 fp32): lanes 0-15 K={k,k+1}, lanes 16-31 K={k+2,k+3}
        v2f a0 = *(const v2f*)(pA + 0);
        v2f b0 = *(const v2f*)(pB + 0);
        acc0 = __builtin_amdgcn_wmma_f32_16x16x4_f32(
                   false, a0, false, b0, (short)0, acc0, false, false);
        v2f a1 = *(const v2f*)(pA + 4);
        v2f b1 = *(const v2f*)(pB + 4);
        acc1 = __builtin_amdgcn_wmma_f32_16x16x4_f32(
                   false, a1, false, b1, (short)0, acc1, false, false);
        v2f a2 = *(const v2f*)(pA + 8);
        v2f b2 = *(const v2f*)(pB + 8);
        acc2 = __builtin_amdgcn_wmma_f32_16x16x4_f32(
                   false, a2, false, b2, (short)0, acc2, false, false);
        v2f a3 = *(const v2f*)(pA + 12);
        v2f b3 = *(const v2f*)(pB + 12);
        acc3 = __builtin_amdgcn_wmma_f32_16x16x4_f32(
                   false, a3, false, b3, (short)0, acc3, false, false);
        pA += 16;
        pB += 16;
    }
    acc0 += acc1;
    acc2 += acc3;
    acc0 += acc2;

    // C layout: VGPR v -> M = v (lanes 0-15) / v+8 (lanes 16-31), N = lane%16
    const int cn = n0 + lrow;
    const int mBase = m0 + half * 8;
#pragma unroll
    for (int v = 0; v < 8; ++v) {
        unsafeAtomicAdd(&gram[(mBase + v) * PADN + cn], acc0[v]);
    }
}

// -------- kernel 3: contrastive loss from gram + inv_norms --------
__global__ void loss_kernel(const float* __restrict__ gram,
                            const float* __restrict__ inv_norm,
                            float* __restrict__ out) {
    __shared__ float red[128];
    const int t = threadIdx.x;
    float lp = 0.0f;
    if (t < NROWS) {
        const float invr = inv_norm[t];
        const int g  = (t % BATCH) / GROUP;
        const int a0 = g * GROUP;                // nominator cols [a0,a0+5) and +BATCH
        const int pc = (t < BATCH) ? (t + BATCH) : (t - BATCH);  // positive column
        float denom = 0.0f, nom = 0.0f, ediag = 0.0f, epos = 0.0f;
        for (int c = 0; c < NROWS; ++c) {
            float s = gram[t * PADN + c] * invr * inv_norm[c];
            float e = expf(s * 2.0f);            // sim / TEMP, TEMP = 0.5
            denom += e;
            if (c >= a0 && c < a0 + GROUP) nom += e;
            if (c >= a0 + BATCH && c < a0 + BATCH + GROUP) nom += e;
            if (c == t)  ediag = e;
            if (c == pc) epos  = e;
        }
        const float denom_sum = denom - ediag;
        const float nominator = nom - ediag;
        // -log(epos / (denom_sum - nominator + epos))
        lp = logf((denom_sum - nominator + epos) / epos);
    }
    red[t] = lp;
    __syncthreads();
    for (int s = 64; s > 0; s >>= 1) {
        if (t < s) red[t] += red[t + s];
        __syncthreads();
    }
    if (t == 0) out[0] = red[0] / 91.0f;
}

extern "C" void kernel_launch(void* const* d_in, const int* in_sizes, int n_in,
                              void* d_out, int out_size, void* d_ws, size_t ws_size,
                              hipStream_t stream) {
    const float* ei = (const float*)d_in[0];
    const float* ej = (const float*)d_in[1];
    float* ws       = (float*)d_ws;
    float* inv_norm = ws;            // 100 floats (128 reserved)
    float* gram     = ws + 128;      // 112*112 floats

    zero_gram_kernel<<<dim3((PADN * PADN + 255) / 256), 256, 0, stream>>>(gram);
    norms_kernel<<<dim3(NROWS), 256, 0, stream>>>(ei, ej, inv_norm);
    gram_kernel<<<dim3(NT * NT, SPLITK), 256, 0, stream>>>(ei, ej, gram);
    loss_kernel<<<dim3(1), 128, 0, stream>>>(gram, inv_norm, (float*)d_out);
}